// SapaFeatureAlignment_71047349010863
// MI455X (gfx1250) — compile-verified
//
#include <hip/hip_runtime.h>
#include <hip/hip_bf16.h>

// SAPA feature alignment for MI455X (gfx1250, wave32, WMMA).
// Matrix math: V_WMMA_F32_16X16X4_F32 (f32 in, f32 accum).
// GEMM staging: GLOBAL_LOAD_ASYNC_TO_LDS_B128 with double-buffered LDS
// (ASYNCcnt-tracked, overlaps the A/B stream with WMMA compute).
//
// Scratch layout (floats), total ~159.4 MB:
//   fine_n   65536*256
//   coarse_n 16384*512
//   q_fine   65536*64
//   q_coarse 16384*64
//   k_       16384*64
//   v_       16384*256
//   gate     16384
//   query    65536*64

typedef __attribute__((ext_vector_type(2))) float v2f;
typedef __attribute__((ext_vector_type(8))) float v8f;

__device__ __forceinline__ v8f wmma_f32(v2f a, v2f b, v8f c) {
    // D = A(16x4) * B(4x16) + C(16x16), all f32
    return __builtin_amdgcn_wmma_f32_16x16x4_f32(
        /*neg_a=*/false, a, /*neg_b=*/false, b,
        /*c_mod=*/(short)0, c, /*reuse_a=*/false, /*reuse_b=*/false);
}

// Async copy 16B global -> LDS (per-lane addresses), tracked by ASYNCcnt.
__device__ __forceinline__ void async_load_b128(void* lds_ptr, const float* gptr) {
    const unsigned lds_addr = (unsigned)(uintptr_t)lds_ptr;       // low 32b = LDS byte offset
    const unsigned long long ga = (unsigned long long)(uintptr_t)gptr;
    asm volatile("global_load_async_to_lds_b128 %0, %1, off"
                 :: "v"(lds_addr), "v"(ga)
                 : "memory");
}
__device__ __forceinline__ void wait_async0() {
    asm volatile("s_wait_asynccnt 0x0" ::: "memory");
}

__device__ __forceinline__ float wave_sum(float s) {
    #pragma unroll
    for (int off = 16; off; off >>= 1) s += __shfl_xor(s, off, 32);
    return s;
}
__device__ __forceinline__ float wave_max(float s) {
    #pragma unroll
    for (int off = 16; off; off >>= 1) s = fmaxf(s, __shfl_xor(s, off, 32));
    return s;
}

// ---------------- LayerNorm: one wave per row ----------------
template <int C>
__global__ __launch_bounds__(256) void ln_kernel(
    const float* __restrict__ x, const float* __restrict__ g,
    const float* __restrict__ b, float* __restrict__ y, int nrows)
{
    const int wv = threadIdx.x >> 5, lane = threadIdx.x & 31;
    const int row = blockIdx.x * 8 + wv;
    if (row >= nrows) return;
    const float* xr = x + (size_t)row * C;
    float* yr = y + (size_t)row * C;
    constexpr int PER = C / 32;
    float v[PER];
    float s = 0.f;
    #pragma unroll
    for (int i = 0; i < PER; ++i) { v[i] = xr[lane + i * 32]; s += v[i]; }
    const float mean = wave_sum(s) * (1.0f / C);
    float vs = 0.f;
    #pragma unroll
    for (int i = 0; i < PER; ++i) { float d = v[i] - mean; vs += d * d; }
    const float rstd = rsqrtf(wave_sum(vs) * (1.0f / C) + 1e-5f);
    #pragma unroll
    for (int i = 0; i < PER; ++i) {
        int c = lane + i * 32;
        yr[c] = (v[i] - mean) * rstd * g[c] + b[c];
    }
}

// ---------------- Generic WMMA GEMM: C[M,N] = A[M,K] @ W[K,N] + bias ----------------
// Block tile 64x64, 256 threads = 8 waves, wave -> 2 (16x16) tiles sharing A-frag.
// Double-buffered LDS filled by GLOBAL_LOAD_ASYNC_TO_LDS_B128.
// M, N multiples of 64; K multiple of 32.
__global__ __launch_bounds__(256) void gemm_kernel(
    const float* __restrict__ A, const float* __restrict__ W,
    const float* __restrict__ bias, float* __restrict__ C,
    int M, int N, int K)
{
    // Row strides are multiples of 16B (36*4=144, 68*4=272) so every B128 write
    // stays inside one padded row. Padding keeps 16-row column reads conflict-free.
    __shared__ __align__(16) float As[2][64][36];
    __shared__ __align__(16) float Bs[2][32][68];

    const int wv = threadIdx.x >> 5, lane = threadIdx.x & 31;
    const int m0 = blockIdx.x * 64, n0 = blockIdx.y * 64;
    const int mt = wv >> 1;            // 0..3
    const int nt0 = (wv & 1) * 2;      // 0 or 2
    const int l15 = lane & 15;
    const int khalf = (lane >> 4) << 1;  // 0 for lanes 0-15, 2 for lanes 16-31

    v8f acc[2] = {};

    // A chunk: 64x32 floats = 512 16B-chunks; B chunk: 32x64 floats = 512 16B-chunks.
    // 256 threads -> 2 chunks each per matrix, issued asynchronously.
    auto issue_chunk = [&](int buf, int k0) {
        #pragma unroll
        for (int i = threadIdx.x; i < 512; i += 256) {
            const int m = i >> 3, kc = (i & 7) * 4;
            async_load_b128(&As[buf][m][kc], &A[(size_t)(m0 + m) * K + k0 + kc]);
        }
        #pragma unroll
        for (int i = threadIdx.x; i < 512; i += 256) {
            const int kc = i >> 4, n = (i & 15) * 4;
            async_load_b128(&Bs[buf][kc][n], &W[(size_t)(k0 + kc) * N + n0 + n]);
        }
    };

    issue_chunk(0, 0);
    wait_async0();
    __syncthreads();

    const int nchunks = K >> 5;
    for (int t = 0; t < nchunks; ++t) {
        const int cur = t & 1;
        if (t + 1 < nchunks) issue_chunk(cur ^ 1, (t + 1) * 32);

        #pragma unroll
        for (int kc = 0; kc < 32; kc += 4) {
            const int kb = kc + khalf;
            v2f a;
            a.x = As[cur][mt * 16 + l15][kb];
            a.y = As[cur][mt * 16 + l15][kb + 1];
            #pragma unroll
            for (int u = 0; u < 2; ++u) {
                const int nt = nt0 + u;
                v2f bf;
                bf.x = Bs[cur][kb][nt * 16 + l15];
                bf.y = Bs[cur][kb + 1][nt * 16 + l15];
                acc[u] = wmma_f32(a, bf, acc[u]);
            }
        }

        if (t + 1 < nchunks) {
            wait_async0();     // own async loads into buf^1 landed
            __syncthreads();   // everyone done reading cur & writing buf^1
        }
    }

    // Store D (+bias). C/D layout: vgpr v -> M = mt*16 + v + (lane>=16 ? 8 : 0), N = lane&15.
    const int rbase = mt * 16 + ((lane >> 4) << 3);
    #pragma unroll
    for (int u = 0; u < 2; ++u) {
        const int col = (nt0 + u) * 16 + l15;
        const float bb = bias[n0 + col];
        #pragma unroll
        for (int vz = 0; vz < 8; ++vz) {
            C[(size_t)(m0 + rbase + vz) * N + n0 + col] = acc[u][vz] + bb;
        }
    }
}

// ---------------- Gate: sigmoid(coarse_n @ wg + bg), wave per row ----------------
__global__ __launch_bounds__(256) void gate_kernel(
    const float* __restrict__ cn, const float* __restrict__ wg,
    const float* __restrict__ bg, float* __restrict__ gate, int nrows)
{
    const int wv = threadIdx.x >> 5, lane = threadIdx.x & 31;
    const int row = blockIdx.x * 8 + wv;
    if (row >= nrows) return;
    const float* xr = cn + (size_t)row * 512;
    float s = 0.f;
    #pragma unroll
    for (int i = 0; i < 16; ++i) s += xr[lane + i * 32] * wg[lane + i * 32];
    s = wave_sum(s);
    if (lane == 0) gate[row] = 1.0f / (1.0f + __expf(-(s + bg[0])));
}

// ---------------- Query blend: q = qf*g + qc_up*(1-g) ----------------
__global__ __launch_bounds__(256) void blend_kernel(
    const float* __restrict__ qf, const float* __restrict__ qc,
    const float* __restrict__ gate, float* __restrict__ query)
{
    const int idx = blockIdx.x * 256 + threadIdx.x;  // total 65536*64
    const int p = idx >> 6, e = idx & 63;
    const int x = p & 127, y = (p >> 7) & 127, b = p >> 14;
    const int cidx = (b * 64 + (y >> 1)) * 64 + (x >> 1);
    const float g = gate[cidx];
    query[idx] = qf[idx] * g + qc[(size_t)cidx * 64 + e] * (1.0f - g);
}

// ---------------- Local attention over a 4x4 coarse tile ----------------
// 64 queries (16 cells x 2x2 fine pixels) vs shared 8x8 key union.
// S = Q*K^T (WMMA, 64x64x64), masked softmax over 5x5 window, O = P*V (WMMA, 64x64 @ 64x256).
__global__ __launch_bounds__(256) void attn_kernel(
    const float* __restrict__ query,  // [B,128,128,64]
    const float* __restrict__ Kg,     // [B,64,64,64]
    const float* __restrict__ Vg,     // [B,64,64,256]
    float* __restrict__ out)          // [B,128,128,256]
{
    __shared__ float Qs[64][68];
    __shared__ float Ks[64][68];
    __shared__ float Ps[64][68];
    __shared__ int   voffs[64];  // V row base offsets (elements) or -1 if zero-padded
    __shared__ int   qpix[64];   // fine-pixel index of each query row

    const int b = blockIdx.z;
    const int i0 = blockIdx.y * 4, j0 = blockIdx.x * 4;
    const int wv = threadIdx.x >> 5, lane = threadIdx.x & 31;
    const int l15 = lane & 15;
    const int khalf = (lane >> 4) << 1;

    // Tables: union position validity/offsets + query-row pixel indices.
    if (threadIdx.x < 64) {
        const int u = threadIdx.x;
        const int r = u >> 3, c = u & 7;
        const int ir = i0 - 2 + r, jc = j0 - 2 + c;
        const bool ok = (ir >= 0) && (ir < 64) && (jc >= 0) && (jc < 64);
        voffs[u] = ok ? (((b * 64 + ir) * 64 + jc) * 256) : -1;
        const int cell = u >> 2, sub = u & 3;
        const int ci = cell >> 2, cj = cell & 3;
        const int y = (i0 + ci) * 2 + (sub >> 1);
        const int x = (j0 + cj) * 2 + (sub & 1);
        qpix[u] = (b * 128 + y) * 128 + x;
    }
    __syncthreads();

    // Stage K union [64,64] with zero padding, and Q [64,64].
    for (int i = threadIdx.x; i < 4096; i += 256) {
        const int u = i >> 6, e = i & 63;
        const int r = u >> 3, c = u & 7;
        const int ir = i0 - 2 + r, jc = j0 - 2 + c;
        const bool ok = (ir >= 0) && (ir < 64) && (jc >= 0) && (jc < 64);
        Ks[u][e] = ok ? Kg[(size_t)((b * 64 + ir) * 64 + jc) * 64 + e] : 0.0f;
        Qs[u][e] = query[(size_t)qpix[u] * 64 + e];
    }
    __syncthreads();

    // ---- S = Q * K^T : M=64, N=64, K=64 (8 waves x 2 tiles) ----
    {
        const int mt = wv >> 1;
        const int nt0 = (wv & 1) * 2;
        v8f accS[2] = {};
        #pragma unroll
        for (int kc = 0; kc < 64; kc += 4) {
            const int kb = kc + khalf;
            v2f a;
            a.x = Qs[mt * 16 + l15][kb];
            a.y = Qs[mt * 16 + l15][kb + 1];
            #pragma unroll
            for (int u = 0; u < 2; ++u) {
                const int nrow = (nt0 + u) * 16 + l15;  // B[k][n] = Ks[n][k]
                v2f bf;
                bf.x = Ks[nrow][kb];
                bf.y = Ks[nrow][kb + 1];
                accS[u] = wmma_f32(a, bf, accS[u]);
            }
        }
        const int rbase = mt * 16 + ((lane >> 4) << 3);
        #pragma unroll
        for (int u = 0; u < 2; ++u) {
            const int col = (nt0 + u) * 16 + l15;
            #pragma unroll
            for (int vz = 0; vz < 8; ++vz) Ps[rbase + vz][col] = accS[u][vz];
        }
    }
    __syncthreads();

    // ---- Masked softmax per query row (wave handles 8 rows; lane covers cols {lane, lane+32}) ----
    #pragma unroll
    for (int rr = 0; rr < 8; ++rr) {
        const int row = wv * 8 + rr;
        const int cell = row >> 2;
        const int ci = cell >> 2, cj = cell & 3;
        const int u0 = lane, u1 = lane + 32;
        const int r0 = u0 >> 3, c0 = u0 & 7;
        const int r1 = u1 >> 3, c1 = u1 & 7;
        const bool v0 = (r0 >= ci) && (r0 <= ci + 4) && (c0 >= cj) && (c0 <= cj + 4);
        const bool v1 = (r1 >= ci) && (r1 <= ci + 4) && (c1 >= cj) && (c1 <= cj + 4);
        const float NEG = -3.0e38f;
        const float x0 = v0 ? Ps[row][u0] : NEG;
        const float x1 = v1 ? Ps[row][u1] : NEG;
        const float m = wave_max(fmaxf(x0, x1));
        const float e0 = v0 ? __expf(x0 - m) : 0.0f;
        const float e1 = v1 ? __expf(x1 - m) : 0.0f;
        const float inv = 1.0f / wave_sum(e0 + e1);
        Ps[row][u0] = e0 * inv;
        Ps[row][u1] = e1 * inv;
    }
    __syncthreads();

    // ---- O = P * V : M=64, K=64, N=256 (8 waves x 8 tiles); V frags straight from global ----
    const int mtO = wv >> 1;
    const int ntO0 = (wv & 1) * 8;
    const int rbase = mtO * 16 + ((lane >> 4) << 3);
    #pragma unroll
    for (int tt = 0; tt < 8; ++tt) {
        const int ncol = (ntO0 + tt) * 16 + l15;
        v8f acc = {};
        #pragma unroll
        for (int kc = 0; kc < 64; kc += 4) {
            const int kb = kc + khalf;
            v2f a;
            a.x = Ps[mtO * 16 + l15][kb];
            a.y = Ps[mtO * 16 + l15][kb + 1];
            const int o0 = voffs[kb], o1 = voffs[kb + 1];
            v2f bf;
            bf.x = (o0 >= 0) ? Vg[(size_t)o0 + ncol] : 0.0f;
            bf.y = (o1 >= 0) ? Vg[(size_t)o1 + ncol] : 0.0f;
            acc = wmma_f32(a, bf, acc);
        }
        #pragma unroll
        for (int vz = 0; vz < 8; ++vz) {
            out[(size_t)qpix[rbase + vz] * 256 + ncol] = acc[vz];
        }
    }
}

// ---------------- Host launcher ----------------
extern "C" void kernel_launch(void* const* d_in, const int* in_sizes, int n_in,
                              void* d_out, int out_size, void* d_ws, size_t ws_size,
                              hipStream_t stream)
{
    const float* fine    = (const float*)d_in[0];   // [4,128,128,256]
    const float* coarse  = (const float*)d_in[1];   // [4,64,64,512]
    const float* gamma_f = (const float*)d_in[2];
    const float* beta_f  = (const float*)d_in[3];
    const float* gamma_c = (const float*)d_in[4];
    const float* beta_c  = (const float*)d_in[5];
    const float* wg      = (const float*)d_in[6];   // [512,1]
    const float* bg      = (const float*)d_in[7];   // [1]
    const float* wqf     = (const float*)d_in[8];   // [256,64]
    const float* bqf     = (const float*)d_in[9];
    const float* wqc     = (const float*)d_in[10];  // [512,64]
    const float* bqc     = (const float*)d_in[11];
    const float* wk      = (const float*)d_in[12];  // [512,64]
    const float* bk      = (const float*)d_in[13];
    const float* wv      = (const float*)d_in[14];  // [512,256]
    const float* bv      = (const float*)d_in[15];
    float* out = (float*)d_out;                     // [4,128,128,256]

    const int MF = 4 * 128 * 128;   // 65536 fine rows
    const int MC = 4 * 64 * 64;     // 16384 coarse rows

    float* ws = (float*)d_ws;
    float* fine_n   = ws;                    // 16777216
    float* coarse_n = fine_n + 16777216;     // 8388608
    float* q_fine   = coarse_n + 8388608;    // 4194304
    float* q_coarse = q_fine + 4194304;      // 1048576
    float* k_       = q_coarse + 1048576;    // 1048576
    float* v_       = k_ + 1048576;          // 4194304
    float* gate     = v_ + 4194304;          // 16384
    float* query    = gate + 16384;          // 4194304

    // 1) LayerNorms
    ln_kernel<256><<<MF / 8, 256, 0, stream>>>(fine, gamma_f, beta_f, fine_n, MF);
    ln_kernel<512><<<MC / 8, 256, 0, stream>>>(coarse, gamma_c, beta_c, coarse_n, MC);

    // 2) Projections (WMMA GEMMs with async double-buffered staging)
    gemm_kernel<<<dim3(MF / 64, 64 / 64), 256, 0, stream>>>(fine_n,   wqf, bqf, q_fine,   MF, 64,  256);
    gemm_kernel<<<dim3(MC / 64, 64 / 64), 256, 0, stream>>>(coarse_n, wqc, bqc, q_coarse, MC, 64,  512);
    gemm_kernel<<<dim3(MC / 64, 64 / 64), 256, 0, stream>>>(coarse_n, wk,  bk,  k_,       MC, 64,  512);
    gemm_kernel<<<dim3(MC / 64, 256 / 64), 256, 0, stream>>>(coarse_n, wv, bv,  v_,       MC, 256, 512);

    // 3) Gate + blended query
    gate_kernel<<<MC / 8, 256, 0, stream>>>(coarse_n, wg, bg, gate, MC);
    blend_kernel<<<(MF * 64) / 256, 256, 0, stream>>>(q_fine, q_coarse, gate, query);

    // 4) Local attention (one block per 4x4 coarse tile)
    attn_kernel<<<dim3(16, 16, 4), 256, 0, stream>>>(query, k_, v_, out);
}